// MultiLobeSGGX_66391604462167
// MI455X (gfx1250) — compile-verified
//
#include <hip/hip_runtime.h>
#include <math.h>

// ---------------------------------------------------------------------------
// MultiLobeSGGX sampling: elementwise, no GEMM structure => WMMA inapplicable.
// Roofline: 72 B/elem @ 23.3 TB/s => ~13 us floor; round-1 build had 384 VALU
// ops/elem (sincosf lib expansion + IEEE div sequences) which makes the kernel
// VALU-bound. This round routes transcendentals to the CDNA5 hardware pipe
// (v_sin/v_cos/v_sqrt/v_rsq/v_rcp) to push compute below the memory floor.
// Memory path: b96/b64/b32 coalesced loads, single b96 non-temporal store.
// ---------------------------------------------------------------------------

struct F3 { float x, y, z; };
struct F2 { float x, y; };

__device__ __forceinline__ float dot3(const F3 a, const F3 b) {
    return fmaf(a.x, b.x, fmaf(a.y, b.y, a.z * b.z));
}

// Duff et al. branchless orthonormal basis (matches reference onb()).
// Uses v_rcp_f32 (~1 ULP) instead of the IEEE div sequence.
__device__ __forceinline__ void onb(const F3 n, F3& t1, F3& t2) {
    float sg = (n.z >= 0.0f) ? 1.0f : -1.0f;
    float a  = -__builtin_amdgcn_rcpf(sg + n.z);
    float b  = n.x * n.y * a;
    t1.x = 1.0f + sg * n.x * n.x * a;
    t1.y = sg * b;
    t1.z = -sg * n.x;
    t2.x = b;
    t2.y = sg + n.y * n.y * a;
    t2.z = -n.y;
}

// sggx_dot(a,b,n,alpha2) = alpha2*dot(a,b) + (1-alpha2)*dot(a,n)*dot(b,n)
__device__ __forceinline__ float sggx_dot(const F3 a, const F3 b, const F3 n,
                                          float alpha2) {
    return alpha2 * dot3(a, b) + (1.0f - alpha2) * dot3(a, n) * dot3(b, n);
}

__global__ void __launch_bounds__(256)
multilobe_sggx_kernel(const float* __restrict__ wi_p,
                      const float* __restrict__ n_p,
                      const float* __restrict__ alb_p,
                      const float* __restrict__ met_p,
                      const float* __restrict__ alpha_p,
                      const float* __restrict__ smp_p,
                      float* __restrict__ out_p,
                      int N) {
    int i = blockIdx.x * blockDim.x + threadIdx.x;
    if (i >= N) return;

    const float EPS = 1e-6f;

    // ---- coalesced wide loads: b96 / b64 / b32 per lane -------------------
    const F3 wi  = reinterpret_cast<const F3*>(wi_p)[i];
    const F3 n   = reinterpret_cast<const F3*>(n_p)[i];
    const F3 alb = reinterpret_cast<const F3*>(alb_p)[i];
    const F3 met = reinterpret_cast<const F3*>(met_p)[i];
    const float alpha = alpha_p[i];
    const F2 smp = reinterpret_cast<const F2*>(smp_p)[i];
    const float u0 = smp.x;
    const float u1 = smp.y;

    // ---- lobe selection probability ---------------------------------------
    const float w_d = fmaf(0.2126f, alb.x, fmaf(0.7152f, alb.y, 0.0722f * alb.z));
    const float w_s = fmaf(0.2126f, met.x, fmaf(0.7152f, met.y, 0.0722f * met.z));
    const float tot = w_d + w_s;
    // Keep IEEE division here: p_spec feeds the discrete lobe-select compare.
    const float p_spec = (tot > EPS) ? (w_s / (tot + EPS)) : 0.0f;
    const bool  spec_mask = p_spec > u0;

    // Remapped u0 per lobe (continuous consumers -> hw rcp is fine).
    const float u0_spec = fminf(fmaxf(u0 * __builtin_amdgcn_rcpf(fmaxf(p_spec, EPS)),
                                      0.0f), 1.0f);
    const float u0_diff = fminf(fmaxf((u0 - p_spec) *
                                      __builtin_amdgcn_rcpf(fmaxf(1.0f - p_spec, EPS)),
                                      0.0f), 1.0f);

    // phi = 2*pi*u1 shared by BOTH lobes; arg in [0, 2pi] -> hardware sin/cos.
    const float phi  = 6.28318530717958647692f * u1;
    const float sphi = __builtin_amdgcn_sinf(phi);
    const float cphi = __builtin_amdgcn_cosf(phi);

    // ---- diffuse lobe: uniform hemisphere ---------------------------------
    F3 t1, t2;
    onb(n, t1, t2);
    const float z = u0_diff;
    const float r = __builtin_amdgcn_sqrtf(fmaxf(0.0f, 1.0f - z * z));
    const float rc = r * cphi, rs = r * sphi;
    F3 wo_diff;
    wo_diff.x = fmaf(t1.x, rc, fmaf(t2.x, rs, n.x * z));
    wo_diff.y = fmaf(t1.y, rc, fmaf(t2.y, rs, n.y * z));
    wo_diff.z = fmaf(t1.z, rc, fmaf(t2.z, rs, n.z * z));

    // ---- specular lobe: SGGX VNDF sampling --------------------------------
    const float a2 = alpha * alpha;
    F3 wk, wj;
    onb(wi, wk, wj);

    const float S_kk = sggx_dot(wk, wk, n, a2);
    const float S_kj = sggx_dot(wk, wj, n, a2);
    const float S_ki = sggx_dot(wk, wi, n, a2);
    const float S_jj = sggx_dot(wj, wj, n, a2);
    const float S_ji = sggx_dot(wj, wi, n, a2);
    const float S_ii = sggx_dot(wi, wi, n, a2);

    const float det = S_kk * S_jj * S_ii - S_kj * S_kj * S_ii
                    - S_ki * S_ki * S_jj - S_ji * S_ji * S_kk
                    + 2.0f * S_kj * S_ki * S_ji;
    const float sqrt_det     = __builtin_amdgcn_sqrtf(fabsf(det));
    const float inv_sqrt_Sii = __builtin_amdgcn_rsqf(S_ii);   // v_rsq_f32
    const float tmp = __builtin_amdgcn_sqrtf(fmaxf(S_jj * S_ii - S_ji * S_ji,
                                                   1e-12f));
    const float inv_tmp = __builtin_amdgcn_rcpf(tmp);

    // M columns (kji frame)
    const float Mk_x = sqrt_det * inv_tmp;
    const float Mj_x = -inv_sqrt_Sii * (S_ki * S_ji - S_kj * S_ii) * inv_tmp;
    const float Mj_y = inv_sqrt_Sii * tmp;
    const float Mi_x = inv_sqrt_Sii * S_ki;
    const float Mi_y = inv_sqrt_Sii * S_ji;
    const float Mi_z = inv_sqrt_Sii * S_ii;

    const float su = __builtin_amdgcn_sqrtf(u0_spec);
    const float uu = su * cphi;
    const float vv = su * sphi;
    const float ww = __builtin_amdgcn_sqrtf(fmaxf(0.0f, 1.0f - u0_spec));

    float wm_k = fmaf(uu, Mk_x, fmaf(vv, Mj_x, ww * Mi_x));
    float wm_j = fmaf(vv, Mj_y, ww * Mi_y);
    float wm_i = ww * Mi_z;
    const float inv_norm =
        __builtin_amdgcn_rsqf(fmaf(wm_k, wm_k, fmaf(wm_j, wm_j, wm_i * wm_i)));
    wm_k *= inv_norm;
    wm_j *= inv_norm;
    wm_i *= inv_norm;

    F3 wm;
    wm.x = fmaf(wm_k, wk.x, fmaf(wm_j, wj.x, wm_i * wi.x));
    wm.y = fmaf(wm_k, wk.y, fmaf(wm_j, wj.y, wm_i * wi.y));
    wm.z = fmaf(wm_k, wk.z, fmaf(wm_j, wj.z, wm_i * wi.z));

    const float two_dp = 2.0f * dot3(wm, wi);
    F3 wo_spec;
    wo_spec.x = fmaf(two_dp, wm.x, -wi.x);
    wo_spec.y = fmaf(two_dp, wm.y, -wi.y);
    wo_spec.z = fmaf(two_dp, wm.z, -wi.z);

    // ---- branchless select + non-temporal store (merges into b96 NT) ------
    const float ox = spec_mask ? wo_spec.x : wo_diff.x;
    const float oy = spec_mask ? wo_spec.y : wo_diff.y;
    const float oz = spec_mask ? wo_spec.z : wo_diff.z;
    float* o = out_p + 3ll * i;
    __builtin_nontemporal_store(ox, o + 0);
    __builtin_nontemporal_store(oy, o + 1);
    __builtin_nontemporal_store(oz, o + 2);
}

extern "C" void kernel_launch(void* const* d_in, const int* in_sizes, int n_in,
                              void* d_out, int out_size, void* d_ws,
                              size_t ws_size, hipStream_t stream) {
    // setup_inputs order: wi, n, albedo, metallic, alpha_x, alpha_y, sample
    const float* wi       = (const float*)d_in[0];
    const float* n        = (const float*)d_in[1];
    const float* albedo   = (const float*)d_in[2];
    const float* metallic = (const float*)d_in[3];
    const float* alpha_x  = (const float*)d_in[4];
    // d_in[5] (alpha_y) is unused by the reference -> never touch it (saves BW)
    const float* sample   = (const float*)d_in[6];
    float* out = (float*)d_out;

    const int N = in_sizes[0] / 3;  // wi has N*3 floats
    const int block = 256;          // 8 wave32s per block on CDNA5
    const int grid  = (N + block - 1) / block;

    multilobe_sggx_kernel<<<grid, block, 0, stream>>>(
        wi, n, albedo, metallic, alpha_x, sample, out, N);
}